// SubGraph_5738076307803
// MI455X (gfx1250) — compile-verified
//
#include <hip/hip_runtime.h>

typedef float v2f __attribute__((ext_vector_type(2)));
typedef float v8f __attribute__((ext_vector_type(8)));

#define NSEG 64
#define HDIM 64
#define BATCH 128
#define NPTS 2048
#define ROWS_PER_BLOCK 512   // 8 waves x 16 rows x 4 m-tiles

__global__ void zero_kernel(float* __restrict__ p, int n) {
    int i = blockIdx.x * blockDim.x + threadIdx.x;
    if (i < n) p[i] = 0.0f;
}

// Butterfly add via ds_swizzle_b32 (group-of-32 mode: offset = xor<<10 | and_mask 0x1f).
// Immediate-encoded, no address VGPR, no bpermute preamble.
template <int MASK>
__device__ __forceinline__ float xadd(float v) {
    int o = __builtin_amdgcn_ds_swizzle(__float_as_int(v), (MASK << 10) | 0x1f);
    return v + __int_as_float(o);
}

// One wave computes 16-row x 64-col tiles of h = relu(LN(x_in @ W + b) * g + beta),
// writes h (optional), segment-maxes into an LDS tile (ds_max_u32 via atomicMax on
// unsigned bits -- valid for post-ReLU values >= 0), then the block flushes the
// LDS tile to agg_out with one global atomic per element.
// FUSED: input channels 0..63 come from h_prev, 64..127 from agg_in[b, cluster[row], :].
template <int K_IN, bool FUSED>
__global__ __launch_bounds__(256) void layer_kernel(
    const float* __restrict__ xin,      // FUSED ? h_prev (B,N,64) : x (B,N,K_IN)
    const float* __restrict__ agg_in,   // (B,64,64) when FUSED, else unused
    const int*   __restrict__ cluster,  // (B,N)
    const float* __restrict__ W,        // (K_IN,64) row-major
    const float* __restrict__ bias,
    const float* __restrict__ gam,
    const float* __restrict__ bet,
    float* __restrict__ h_out,          // (B,N,64) or nullptr
    float* __restrict__ agg_out)        // (B,64,64), pre-zeroed, uint-max semantics
{
    __shared__ float    sW[K_IN * HDIM];
    __shared__ unsigned sAgg[NSEG * HDIM];   // 16KB block-local segment max (fp32 bits)
    __shared__ float    sB[HDIM];
    __shared__ float    sG[HDIM];
    __shared__ float    sBe[HDIM];

    for (int i = threadIdx.x; i < K_IN * HDIM; i += blockDim.x) sW[i] = W[i];
    for (int i = threadIdx.x; i < NSEG * HDIM; i += blockDim.x) sAgg[i] = 0u;
    if (threadIdx.x < HDIM) {
        sB[threadIdx.x]  = bias[threadIdx.x];
        sG[threadIdx.x]  = gam[threadIdx.x];
        sBe[threadIdx.x] = bet[threadIdx.x];
    }
    __syncthreads();

    const int lane = threadIdx.x & 31;
    const int wave = threadIdx.x >> 5;
    const int b    = blockIdx.y;
    const int hl   = lane & 15;
    const bool hi  = lane >= 16;
    const int koff = hi ? 2 : 0;

    // Per-lane copies of bias / gamma / beta for the 4 column tiles.
    float bn[4], gn[4], betn[4];
    #pragma unroll
    for (int t = 0; t < 4; t++) {
        const int n = t * 16 + hl;
        bn[t] = sB[n]; gn[t] = sG[n]; betn[t] = sBe[n];
    }
    const float bsum  = bn[0] + bn[1] + bn[2] + bn[3];
    const float inv64 = 1.0f / 64.0f;

    for (int mt = 0; mt < ROWS_PER_BLOCK / 128; mt++) {
        const int m0   = blockIdx.x * ROWS_PER_BLOCK + mt * 128 + wave * 16;
        const int mrow = m0 + hl;                    // row this lane supplies for A
        const int cl   = cluster[b * NPTS + mrow];   // cluster of this lane's row

        v8f acc[4];
        #pragma unroll
        for (int t = 0; t < 4; t++) acc[t] = (v8f){};

        #pragma unroll 4
        for (int k0 = 0; k0 < K_IN; k0 += 4) {
            const int kb = k0 + koff;
            v2f a;
            if (FUSED) {
                if (k0 < HDIM) {
                    const float2 t2 = *(const float2*)(xin + (size_t)(b * NPTS + mrow) * HDIM + kb);
                    a.x = t2.x; a.y = t2.y;
                } else {
                    const float2 t2 = *(const float2*)(agg_in + (size_t)(b * NSEG + cl) * HDIM + (kb - HDIM));
                    a.x = t2.x; a.y = t2.y;
                }
            } else {
                const float2 t2 = *(const float2*)(xin + (size_t)(b * NPTS + mrow) * K_IN + kb);
                a.x = t2.x; a.y = t2.y;
            }
            #pragma unroll
            for (int t = 0; t < 4; t++) {
                v2f bf;
                bf.x = sW[(kb)     * HDIM + t * 16 + hl];
                bf.y = sW[(kb + 1) * HDIM + t * 16 + hl];
                acc[t] = __builtin_amdgcn_wmma_f32_16x16x4_f32(
                    false, a, false, bf, (short)0, acc[t], false, false);
            }
        }

        // C/D layout: VGPR r holds row (m0+r) in lanes 0-15 and row (m0+r+8) in lanes
        // 16-31, column = t*16 + (lane&15). LayerNorm reduces across 16 lanes per half.
        // All 8 rows reduced together so the ds_swizzle steps have 8-wide ILP.
        float p[8], mu[8], q[8], rs[8];
        #pragma unroll
        for (int r = 0; r < 8; r++)
            p[r] = acc[0][r] + acc[1][r] + acc[2][r] + acc[3][r] + bsum;
        #pragma unroll
        for (int r = 0; r < 8; r++) p[r] = xadd<1>(p[r]);
        #pragma unroll
        for (int r = 0; r < 8; r++) p[r] = xadd<2>(p[r]);
        #pragma unroll
        for (int r = 0; r < 8; r++) p[r] = xadd<4>(p[r]);
        #pragma unroll
        for (int r = 0; r < 8; r++) p[r] = xadd<8>(p[r]);
        #pragma unroll
        for (int r = 0; r < 8; r++) mu[r] = p[r] * inv64;

        #pragma unroll
        for (int r = 0; r < 8; r++) {
            const float d0 = acc[0][r] + bn[0] - mu[r];
            const float d1 = acc[1][r] + bn[1] - mu[r];
            const float d2 = acc[2][r] + bn[2] - mu[r];
            const float d3 = acc[3][r] + bn[3] - mu[r];
            q[r] = d0 * d0 + d1 * d1 + d2 * d2 + d3 * d3;
        }
        #pragma unroll
        for (int r = 0; r < 8; r++) q[r] = xadd<1>(q[r]);
        #pragma unroll
        for (int r = 0; r < 8; r++) q[r] = xadd<2>(q[r]);
        #pragma unroll
        for (int r = 0; r < 8; r++) q[r] = xadd<4>(q[r]);
        #pragma unroll
        for (int r = 0; r < 8; r++) q[r] = xadd<8>(q[r]);
        #pragma unroll
        for (int r = 0; r < 8; r++) rs[r] = rsqrtf(q[r] * inv64 + 1e-5f);

        int segs[8];
        #pragma unroll
        for (int r = 0; r < 8; r++) segs[r] = __shfl(cl, r + (hi ? 8 : 0), 32);

        #pragma unroll
        for (int r = 0; r < 8; r++) {
            const float o0 = fmaxf((acc[0][r] + bn[0] - mu[r]) * rs[r] * gn[0] + betn[0], 0.0f);
            const float o1 = fmaxf((acc[1][r] + bn[1] - mu[r]) * rs[r] * gn[1] + betn[1], 0.0f);
            const float o2 = fmaxf((acc[2][r] + bn[2] - mu[r]) * rs[r] * gn[2] + betn[2], 0.0f);
            const float o3 = fmaxf((acc[3][r] + bn[3] - mu[r]) * rs[r] * gn[3] + betn[3], 0.0f);

            const int rowm = m0 + r + (hi ? 8 : 0);
            if (h_out != nullptr) {
                float* hp = h_out + (size_t)(b * NPTS + rowm) * HDIM + hl;
                hp[0]  = o0;
                hp[16] = o1;
                hp[32] = o2;
                hp[48] = o3;
            }
            // post-ReLU values are >= 0, so IEEE bits compare like unsigned ints
            unsigned* ap = &sAgg[segs[r] * HDIM + hl];
            atomicMax(ap + 0,  __float_as_uint(o0));
            atomicMax(ap + 16, __float_as_uint(o1));
            atomicMax(ap + 32, __float_as_uint(o2));
            atomicMax(ap + 48, __float_as_uint(o3));
        }
    }

    __syncthreads();

    // Flush block-local segment maxima to global (one atomic per tile element).
    unsigned* gAgg = (unsigned*)(agg_out + (size_t)b * NSEG * HDIM);
    for (int i = threadIdx.x; i < NSEG * HDIM; i += blockDim.x) {
        const unsigned m = sAgg[i];
        if (m != 0u) atomicMax(&gAgg[i], m);
    }
}

// Final output = concat([agg2, agg2], ch-axis), L2-normalized over the segment axis.
__global__ void finalize_kernel(const float* __restrict__ agg, float* __restrict__ out) {
    const int b = blockIdx.x;
    const int c = threadIdx.x;   // 0..63
    float s = 0.0f;
    for (int seg = 0; seg < NSEG; seg++) {
        const float v = agg[(size_t)(b * NSEG + seg) * HDIM + c];
        s += v * v;
    }
    const float inv = 1.0f / fmaxf(sqrtf(s), 1e-12f);
    for (int seg = 0; seg < NSEG; seg++) {
        const float v = agg[(size_t)(b * NSEG + seg) * HDIM + c] * inv;
        out[(size_t)(b * NSEG + seg) * 128 + c]      = v;
        out[(size_t)(b * NSEG + seg) * 128 + c + 64] = v;
    }
}

extern "C" void kernel_launch(void* const* d_in, const int* in_sizes, int n_in,
                              void* d_out, int out_size, void* d_ws, size_t ws_size,
                              hipStream_t stream) {
    const float* x       = (const float*)d_in[0];
    const int*   cluster = (const int*)d_in[1];
    const float* W0 = (const float*)d_in[2];
    const float* b0 = (const float*)d_in[3];
    const float* g0 = (const float*)d_in[4];
    const float* e0 = (const float*)d_in[5];
    const float* W1 = (const float*)d_in[6];
    const float* b1 = (const float*)d_in[7];
    const float* g1 = (const float*)d_in[8];
    const float* e1 = (const float*)d_in[9];
    const float* W2 = (const float*)d_in[10];
    const float* b2 = (const float*)d_in[11];
    const float* g2 = (const float*)d_in[12];
    const float* e2 = (const float*)d_in[13];

    const size_t hElems   = (size_t)BATCH * NPTS * HDIM;   // 16M floats
    const size_t aggElems = (size_t)BATCH * NSEG * HDIM;   // 0.5M floats

    float* ws   = (float*)d_ws;
    float* hA   = ws;
    float* hB   = hA + hElems;
    float* aggA = hB + hElems;
    float* aggB = aggA + aggElems;
    float* aggC = aggB + aggElems;

    const int zn = (int)(3 * aggElems);
    zero_kernel<<<(zn + 255) / 256, 256, 0, stream>>>(aggA, zn);

    dim3 grid(NPTS / ROWS_PER_BLOCK, BATCH);   // (4, 128)
    layer_kernel<8,   false><<<grid, 256, 0, stream>>>(x,  nullptr, cluster, W0, b0, g0, e0, hA, aggA);
    layer_kernel<128, true ><<<grid, 256, 0, stream>>>(hA, aggA,    cluster, W1, b1, g1, e1, hB, aggB);
    layer_kernel<128, true ><<<grid, 256, 0, stream>>>(hB, aggB,    cluster, W2, b2, g2, e2, nullptr, aggC);

    finalize_kernel<<<BATCH, HDIM, 0, stream>>>(aggC, (float*)d_out);
}